// LearnableInterpolation_64037962383646
// MI455X (gfx1250) — compile-verified
//
#include <hip/hip_runtime.h>

// LearnableInterpolation on gfx1250 (MI455X), fused single kernel.
//
// out[b,t,n] = sum_o softmax_o(-3*(tpos[o,n]-tgt[t])^2) * x[b,o,n]
//   tpos[o,n] = (o-L)*ratio[n] + bias[n],  ratio in [0.7,1.3] -> tpos monotone in o.
//
// Gaussian weights have compact support: a K=64 window centered on the analytic
// argmax o* = (tgt-b)/ratio + L reproduces the full softmax exactly in fp32
// (excluded terms are < e^-600 relative to the max -> underflow to 0).
//
// Per workgroup (1 wave32): head n, 16 target columns. GEMM tile:
//   D[32x16] = X[32x64] * W[64x16]  via 32x V_WMMA_F32_16X16X4_F32.
//
// Register layouts (CDNA5 ISA 7.12.2, 32-bit operands, wave32):
//   A 16x4 (MxK):  elem(m,k) -> vgpr (k&1), lane m + 16*(k>=2)
//   B 4x16 (KxN):  elem(k,t) -> vgpr (k&1), lane t + 16*(k>=2)
//   C/D 16x16:     vgpr i: lanes 0-15 -> M=i, lanes 16-31 -> M=i+8; N = lane&15
// => lane's B slots at k-step s are W[4s + 2*(lane>>4) + j][lane&15], j=0,1:
//    each lane computes its own weights straight into the B operand, no LDS.

typedef __attribute__((ext_vector_type(2))) float v2f;
typedef __attribute__((ext_vector_type(8))) float v8f;

#define L_ORG 2048
#define L_TO  2048
#define KWIN  64
#define NSTEP (KWIN / 4)

__global__ __launch_bounds__(32)
void li_wmma_kernel(const float* __restrict__ x,    // [B=32, L_ORG, N]
                    const float* __restrict__ rm1,  // [N]
                    const float* __restrict__ bias, // [N]
                    float* __restrict__ out,        // [B=32, L_TO, N]
                    int N)
{
    const int n    = blockIdx.y;
    const int tt   = blockIdx.x;       // 16-wide tile of target positions
    const int lane = threadIdx.x;      // 0..31 (wave32)
    const int tl   = lane & 15;        // this lane's column within the tile
    const int hi   = lane >> 4;        // half-wave id

    // per-head scalars (clamped as in reference)
    float ratio = fminf(fmaxf(rm1[n] + 1.0f, 0.7f), 1.3f);
    float bb    = fminf(fmaxf(bias[n], -4.0f), 4.0f);
    const float inv_ratio = 1.0f / ratio;

    // this lane's target position
    const int   t_idx = tt * 16 + tl;
    const float tgt   = (float)(t_idx - L_TO);

    // analytic softmax max term: closest o on the monotone tpos line
    float ostar = (tgt - bb) * inv_ratio + (float)L_ORG;
    int o0 = (int)floorf(ostar);
    o0 = min(max(o0, 0), L_ORG - 1);
    int o1 = min(o0 + 1, L_ORG - 1);
    float d0 = (float)(o0 - L_ORG) * ratio + bb - tgt;
    float d1 = (float)(o1 - L_ORG) * ratio + bb - tgt;
    float m  = -3.0f * fminf(d0 * d0, d1 * d1);   // max_o of (-3*delta^2), <= 0

    // uniform window base for the whole tile (covers every lane's o* +- margin)
    float ostar_c = ((float)(tt * 16) + 7.5f - (float)L_TO - bb) * inv_ratio + (float)L_ORG;
    int o_base = (int)ostar_c - (KWIN / 2);
    o_base = min(max(o_base, 0), L_ORG - KWIN);

    // ---- generate W directly in B-operand layout + softmax denominator ----
    const float LOG2E = 1.4426950408889634f;
    v2f  Bv[NSTEP];
    float dsum = 0.0f;
#pragma unroll
    for (int s = 0; s < NSTEP; ++s) {
#pragma unroll
        for (int j = 0; j < 2; ++j) {
            int   k  = 4 * s + 2 * hi + j;
            float tp = (float)(o_base + k - L_ORG) * ratio + bb;
            float dd = tp - tgt;
            float w  = __builtin_amdgcn_exp2f((-3.0f * dd * dd - m) * LOG2E);
            Bv[s][j] = w;
            dsum += w;
        }
    }
    // lanes L and L^16 hold the same column t with disjoint k sets -> one xor-add
    dsum += __shfl_xor(dsum, 16, 32);
    const float inv_denom = 1.0f / dsum;

    // ---- load X window straight into A-operand layout (two M-tiles) ----
    v2f A0[NSTEP], A1[NSTEP];
#pragma unroll
    for (int s = 0; s < NSTEP; ++s) {
        int o = o_base + 4 * s + 2 * hi;
        const float* p0 = x + ((tl)      * L_ORG + o) * N + n;  // rows b=0..15
        const float* p1 = x + ((tl + 16) * L_ORG + o) * N + n;  // rows b=16..31
        A0[s][0] = p0[0];  A0[s][1] = p0[N];
        A1[s][0] = p1[0];  A1[s][1] = p1[N];
    }

    // ---- D[32x16] = X * W : 32x v_wmma_f32_16x16x4_f32 ----
    v8f c0 = {};
    v8f c1 = {};
#pragma unroll
    for (int s = 0; s < NSTEP; ++s) {
        c0 = __builtin_amdgcn_wmma_f32_16x16x4_f32(
                 false, A0[s], false, Bv[s], (short)0, c0, false, false);
        c1 = __builtin_amdgcn_wmma_f32_16x16x4_f32(
                 false, A1[s], false, Bv[s], (short)0, c1, false, false);
    }

    // ---- normalize by softmax denominator and scatter to [B, L_TO, N] ----
#pragma unroll
    for (int i = 0; i < 8; ++i) {
        int brow = i + 8 * hi;                       // D-tile M index for this lane
        out[((brow)      * L_TO + t_idx) * N + n] = c0[i] * inv_denom;
        out[((brow + 16) * L_TO + t_idx) * N + n] = c1[i] * inv_denom;
    }
}

extern "C" void kernel_launch(void* const* d_in, const int* in_sizes, int n_in,
                              void* d_out, int out_size, void* d_ws, size_t ws_size,
                              hipStream_t stream) {
    const float* x    = (const float*)d_in[0];  // org_tensor [32, 2048, 21] f32
    const float* rm1  = (const float*)d_in[1];  // ratio_minus_one [N] f32
    const float* bias = (const float*)d_in[2];  // bias [N] f32
    // d_in[3] is to_length (== 2048, fixed by the harness setup)
    const int N = in_sizes[1];                  // 21

    dim3 grid(L_TO / 16, N);                    // 128 t-tiles x 21 heads
    li_wmma_kernel<<<grid, 32, 0, stream>>>(x, rm1, bias, (float*)d_out, N);
}